// GraphAttention_50113678409872
// MI455X (gfx1250) — compile-verified
//
#include <hip/hip_runtime.h>
#include <hip/hip_bf16.h>

typedef __attribute__((ext_vector_type(2))) float v2f;
typedef __attribute__((ext_vector_type(8))) float v8f;

#define IN_FEAT 128
#define UNITS   64

// ---------------------------------------------------------------------------
// K1: h = X (n_nodes x 128) @ W (128 x 64), pure-f32 WMMA 16x16x4.
// Block = 128 threads = 4 waves; block handles one 16-row M-tile, each wave one
// 16-col N-tile (UNITS = 64 = 4 tiles). K-loop: 128/4 = 32 WMMA issues.
// A layout (32-bit A 16x4): lanes 0-15 -> M=lane, K = k+{0,1}; lanes 16-31 ->
// M=lane-16, K = k+{2,3}. B symmetric (K rows striped across lanes).
// C/D: VGPR v, lane l -> M = v + 8*(l>=16), N = l%16.
// ---------------------------------------------------------------------------
__global__ __launch_bounds__(128)
void gat_gemm_h(const float* __restrict__ X, const float* __restrict__ W,
                float* __restrict__ H, int n_nodes) {
  const int m0   = blockIdx.x * 16;
  const int n0   = (threadIdx.x >> 5) * 16;
  const int lane = threadIdx.x & 31;
  const int mr   = lane & 15;   // row-in-tile for A, col-in-tile for B/C
  const int hi   = lane >> 4;   // 0: K+{0,1}, 1: K+{2,3}

  // Clamp row for tail safety (keeps EXEC all-ones through the WMMAs).
  int arow = m0 + mr; if (arow >= n_nodes) arow = n_nodes - 1;
  const float* __restrict__ xrow = X + (size_t)arow * IN_FEAT;

  v8f c = {};
  #pragma unroll 4
  for (int k = 0; k < IN_FEAT; k += 4) {
    const int ka = k + 2 * hi;
    v2f a, b;
    a.x = xrow[ka + 0];
    a.y = xrow[ka + 1];
    b.x = W[(ka + 0) * UNITS + n0 + mr];
    b.y = W[(ka + 1) * UNITS + n0 + mr];
    c = __builtin_amdgcn_wmma_f32_16x16x4_f32(
        /*neg_a=*/false, a, /*neg_b=*/false, b,
        /*c_mod=*/(short)0, c, /*reuse_a=*/false, /*reuse_b=*/false);
  }

  #pragma unroll
  for (int v = 0; v < 8; ++v) {
    const int m = m0 + v + 8 * hi;
    if (m < n_nodes) H[(size_t)m * UNITS + n0 + mr] = c[v];
  }
}

// ---------------------------------------------------------------------------
// K2: s1[n] = h[n] . a[0:64],  s2[n] = h[n] . a[64:128]. One wave per node.
// ---------------------------------------------------------------------------
__global__ __launch_bounds__(256)
void gat_node_scores(const float* __restrict__ H, const float* __restrict__ KA,
                     float* __restrict__ S1, float* __restrict__ S2, int n_nodes) {
  const int node = blockIdx.x * 8 + (threadIdx.x >> 5);
  const int lane = threadIdx.x & 31;
  if (node >= n_nodes) return;
  const float* __restrict__ hrow = H + (size_t)node * UNITS;
  const float h0 = hrow[lane], h1 = hrow[lane + 32];
  float p1 = h0 * KA[lane]      + h1 * KA[lane + 32];
  float p2 = h0 * KA[64 + lane] + h1 * KA[96 + lane];
  #pragma unroll
  for (int off = 16; off > 0; off >>= 1) {
    p1 += __shfl_down(p1, off, 32);
    p2 += __shfl_down(p2, off, 32);
  }
  if (lane == 0) { S1[node] = p1; S2[node] = p2; }
}

// ---------------------------------------------------------------------------
// K3: CSR row pointers via lower_bound on the sorted dst column.
// ---------------------------------------------------------------------------
__global__ __launch_bounds__(256)
void gat_rowptr(const long long* __restrict__ edges, int* __restrict__ RP,
                int n_nodes, int n_edges) {
  const int n = blockIdx.x * blockDim.x + threadIdx.x;
  if (n > n_nodes) return;
  int lo = 0, hi = n_edges;
  while (lo < hi) {
    const int mid = (lo + hi) >> 1;
    if (edges[2 * (size_t)mid] < (long long)n) lo = mid + 1; else hi = mid;
  }
  RP[n] = lo;
}

// ---------------------------------------------------------------------------
// K4: score[e] = exp(clip(leaky_relu(s1[dst]+s2[src], 0.2), -2, 2))
// ---------------------------------------------------------------------------
__global__ __launch_bounds__(256)
void gat_edge_scores(const long long* __restrict__ edges,
                     const float* __restrict__ S1, const float* __restrict__ S2,
                     float* __restrict__ SC, int n_edges) {
  const int e = blockIdx.x * blockDim.x + threadIdx.x;
  if (e >= n_edges) return;
  const long long d = edges[2 * (size_t)e];
  const long long s = edges[2 * (size_t)e + 1];
  float sc = S1[d] + S2[s];
  sc = (sc > 0.f) ? sc : 0.2f * sc;
  sc = fminf(fmaxf(sc, -2.0f), 2.0f);
  SC[e] = __expf(sc);
}

// ---------------------------------------------------------------------------
// K5: out[n] = sum_e(score * h[src]) / sum_e(score). Wave per node; lane
// covers features {lane, lane+32}. No atomics; h-row gathers are L2-resident
// (25.6 MB << 192 MB L2).
// ---------------------------------------------------------------------------
__global__ __launch_bounds__(256)
void gat_aggregate(const long long* __restrict__ edges,
                   const float* __restrict__ H, const float* __restrict__ SC,
                   const int* __restrict__ RP, float* __restrict__ OUT,
                   int n_nodes) {
  const int node = blockIdx.x * 8 + (threadIdx.x >> 5);
  const int lane = threadIdx.x & 31;
  if (node >= n_nodes) return;
  const int beg = RP[node], end = RP[node + 1];
  float acc0 = 0.f, acc1 = 0.f, ssum = 0.f;
  for (int e = beg; e < end; ++e) {
    const float sc = SC[e];
    const long long s = edges[2 * (size_t)e + 1];
    const float* __restrict__ hs = H + (size_t)s * UNITS;
    acc0 += sc * hs[lane];
    acc1 += sc * hs[lane + 32];
    ssum += sc;
  }
  const float inv = (end > beg) ? 1.0f / ssum : 0.0f;
  OUT[(size_t)node * UNITS + lane]      = acc0 * inv;
  OUT[(size_t)node * UNITS + lane + 32] = acc1 * inv;
}

// ---------------------------------------------------------------------------
extern "C" void kernel_launch(void* const* d_in, const int* in_sizes, int n_in,
                              void* d_out, int out_size, void* d_ws, size_t ws_size,
                              hipStream_t stream) {
  const float*     X  = (const float*)d_in[0];      // (n_nodes, 128)
  const long long* E  = (const long long*)d_in[1];  // (n_edges, 2) int64 [dst,src]
  const float*     W  = (const float*)d_in[2];      // (128, 64)
  const float*     KA = (const float*)d_in[3];      // (128, 1)

  const int n_nodes = in_sizes[0] / IN_FEAT;
  const int n_edges = in_sizes[1] / 2;

  // Workspace carve-up (256-byte aligned slabs).
  char* base = (char*)d_ws;
  size_t off = 0;
  auto carve = [&](size_t bytes) {
    void* p = base + off;
    off += (bytes + 255) & ~(size_t)255;
    return p;
  };
  float* H  = (float*)carve((size_t)n_nodes * UNITS * sizeof(float)); // 25.6 MB
  float* S1 = (float*)carve((size_t)n_nodes * sizeof(float));
  float* S2 = (float*)carve((size_t)n_nodes * sizeof(float));
  int*   RP = (int*)  carve((size_t)(n_nodes + 1) * sizeof(int));
  float* SC = (float*)carve((size_t)n_edges * sizeof(float));         // 6.4 MB
  (void)ws_size;

  const int mtiles = (n_nodes + 15) / 16;
  gat_gemm_h<<<mtiles, 128, 0, stream>>>(X, W, H, n_nodes);

  gat_node_scores<<<(n_nodes + 7) / 8, 256, 0, stream>>>(H, KA, S1, S2, n_nodes);

  gat_rowptr<<<(n_nodes + 256) / 256, 256, 0, stream>>>(E, RP, n_nodes, n_edges);

  gat_edge_scores<<<(n_edges + 255) / 256, 256, 0, stream>>>(E, S1, S2, SC, n_edges);

  gat_aggregate<<<(n_nodes + 7) / 8, 256, 0, stream>>>(E, H, SC, RP, (float*)d_out, n_nodes);
}